// MessagePassingLayer_78039555768697
// MI455X (gfx1250) — compile-verified
//
#include <hip/hip_runtime.h>
#include <stdint.h>

// Problem constants (from the reference): B=8, C=64, N=4096, K=16.
#define BATCH  8
#define CHN    64
#define NNODE  4096
#define KNBR   16

#define NT      256            // nodes per block (== blockDim.x)
#define CH      4              // channel rows staged per TDM chunk (4 x 16KB = 64KB)
#define CPB     16             // channels processed per block
#define NCHUNK  (CPB / CH)     // 4 double-buffered chunks per block

typedef uint32_t u32x4 __attribute__((ext_vector_type(4)));
typedef uint32_t u32x8 __attribute__((ext_vector_type(8)));

// Issue one Tensor-Data-Mover DMA: copy CH rows of NNODE f32 (row stride NNODE
// elems in global) into LDS at byte offset lds_off. Descriptor per
// CDNA5 ISA (§8.3/8.4): 2D tile -> groups 0 and 1 only.
__device__ __forceinline__ void tdm_load_rows(const float* gsrc, uint32_t lds_off) {
  uint64_t ga = (uint64_t)(uintptr_t)gsrc;
  u32x4 g0;
  g0.x = 1u;                                    // count=1, user descriptor
  g0.y = lds_off;                               // lds_addr (bytes)
  g0.z = (uint32_t)ga;                          // global_addr[31:0]
  g0.w = (uint32_t)(ga >> 32) | 0x80000000u;    // global_addr[56:32] | type=2<<30
  u32x8 g1;
  g1[0] = 0x00020000u;                          // workgroup_mask=0, data_size=2 (4B)
  g1[1] = ((uint32_t)NNODE & 0xFFFFu) << 16;    // tensor_dim0[15:0] = 4096
  g1[2] = ((uint32_t)NNODE >> 16)               // tensor_dim0[31:16] = 0
        | ((uint32_t)CH << 16);                 // tensor_dim1[15:0]  = CH
  g1[3] = ((uint32_t)NNODE & 0xFFFFu) << 16;    // tensor_dim1 hi=0 | tile_dim0=4096
  g1[4] = (uint32_t)CH;                         // tile_dim1=CH | tile_dim2=0
  g1[5] = (uint32_t)NNODE;                      // tensor_dim0_stride lo32 = 4096
  g1[6] = 0u;                                   // stride0 hi | dim1_stride lo
  g1[7] = 0u;                                   // dim1_stride hi
  asm volatile("tensor_load_to_lds %0, %1" :: "s"(g0), "s"(g1) : "memory");
}

__global__ void __launch_bounds__(NT) mp_gather_kernel(
    const float* __restrict__ x, const int* __restrict__ ei,
    const float* __restrict__ alphap, float* __restrict__ out) {
  // Double-buffered staging: 2 x CH x 16KB = 128KB of the WGP's 320KB LDS.
  __shared__ float sbuf[2][CH * NNODE];

  const int t  = (int)threadIdx.x;
  const int n0 = (int)blockIdx.x * NT;
  const int b  = (int)blockIdx.y;
  const int c0 = (int)blockIdx.z * CPB;
  const int n  = n0 + t;

  const float scale = 1.0f + alphap[0];   // uniform scalar load

  // 16 neighbor indices per node, contiguous 64B: two global_load_b128 pairs.
  const int4* ep = (const int4*)(ei + ((size_t)b * NNODE + n) * KNBR);
  int4 e0 = ep[0], e1 = ep[1], e2 = ep[2], e3 = ep[3];
  int idxv[KNBR] = {e0.x, e0.y, e0.z, e0.w, e1.x, e1.y, e1.z, e1.w,
                    e2.x, e2.y, e2.z, e2.w, e3.x, e3.y, e3.z, e3.w};

  const float* xb = x   + ((size_t)b * CHN + c0) * NNODE;
  float*       ob = out + ((size_t)b * CHN + c0) * NNODE;

  // Generic LDS pointer low 32 bits == workgroup-relative LDS byte offset.
  const uint32_t lds0 = (uint32_t)(uintptr_t)&sbuf[0][0];
  const uint32_t lds1 = (uint32_t)(uintptr_t)&sbuf[1][0];

  const bool issuer = (t < 32);   // wave 0 issues the per-workgroup DMAs

  if (issuer) {
    tdm_load_rows(xb,              lds0);      // chunk 0 -> buf 0
    tdm_load_rows(xb + CH * NNODE, lds1);      // chunk 1 -> buf 1
  }

  #pragma unroll
  for (int chk = 0; chk < NCHUNK; ++chk) {
    if (issuer) {
      // TENSORcnt completes in order: <=1 outstanding => chunk chk landed.
      if (chk + 1 < NCHUNK) __builtin_amdgcn_s_wait_tensorcnt(1);
      else                  __builtin_amdgcn_s_wait_tensorcnt(0);
    }
    __syncthreads();   // publish staged rows to all 8 waves

    const float* rows = &sbuf[chk & 1][0];

    // k outer / channel inner: the four gathers per neighbor share one base
    // address and sit 16KB (= 64 * 256B) apart -> backend pairs them into
    // ds_load_2addr_stride64_b32 {off0,off1} = {0,64} and {128,192}.
    float acc0 = 0.0f, acc1 = 0.0f, acc2 = 0.0f, acc3 = 0.0f;
    #pragma unroll
    for (int k = 0; k < KNBR; ++k) {
      const float* g = rows + idxv[k];
      acc0 += g[0 * NNODE];
      acc1 += g[1 * NNODE];
      acc2 += g[2 * NNODE];
      acc3 += g[3 * NNODE];
    }
    const float* rn = rows + n;                // own values, same pairing
    float xv0 = rn[0 * NNODE], xv1 = rn[1 * NNODE];
    float xv2 = rn[2 * NNODE], xv3 = rn[3 * NNODE];

    float* o = ob + (size_t)chk * CH * NNODE + n;
    __builtin_nontemporal_store(fmaf(xv0, scale, acc0), o + 0 * NNODE);
    __builtin_nontemporal_store(fmaf(xv1, scale, acc1), o + 1 * NNODE);
    __builtin_nontemporal_store(fmaf(xv2, scale, acc2), o + 2 * NNODE);
    __builtin_nontemporal_store(fmaf(xv3, scale, acc3), o + 3 * NNODE);

    __syncthreads();   // everyone done reading buf[chk&1] before reuse

    if (issuer && (chk + 2 < NCHUNK)) {
      tdm_load_rows(xb + (chk + 2) * CH * NNODE, (chk & 1) ? lds1 : lds0);
    }
  }
}

extern "C" void kernel_launch(void* const* d_in, const int* in_sizes, int n_in,
                              void* d_out, int out_size, void* d_ws, size_t ws_size,
                              hipStream_t stream) {
  (void)in_sizes; (void)n_in; (void)out_size; (void)d_ws; (void)ws_size;
  const float* x  = (const float*)d_in[0];
  const int*   ei = (const int*)d_in[1];
  const float* al = (const float*)d_in[2];
  float*       out = (float*)d_out;

  dim3 grid(NNODE / NT, BATCH, CHN / CPB);   // 16 x 8 x 4 = 512 workgroups
  dim3 block(NT);                            // 256 threads = 8 wave32
  hipLaunchKernelGGL(mp_gather_kernel, grid, block, 0, stream, x, ei, al, out);
}